// SpectralWavelet_90460601188771
// MI455X (gfx1250) — compile-verified
//
#include <hip/hip_runtime.h>
#include <hip/hip_bf16.h>

#define NPAD    512
#define NNODES  500
#define NFEAT   32768
#define NEDGES  16000
#define NSCALES 4
#define TAYLOR_ORDER 8
#define SQUARINGS    6   // B = -(s/64)*L, ||B|| <= 0.125

typedef __attribute__((ext_vector_type(2))) float v2f;
typedef __attribute__((ext_vector_type(8))) float v8f;

__device__ __forceinline__ v8f wmma_f32(v2f a, v2f b, v8f c) {
  // D = A(16x4, f32) * B(4x16, f32) + C(16x16, f32)
  return __builtin_amdgcn_wmma_f32_16x16x4_f32(false, a, false, b, (short)0, c,
                                               false, false);
}

// ---------------- graph construction ----------------

__global__ void k_zero(float* __restrict__ p, int n) {
  int i = blockIdx.x * blockDim.x + threadIdx.x;
  if (i < n) p[i] = 0.0f;
}

__global__ void k_scatter(const int* __restrict__ ei, float* __restrict__ adj,
                          float* __restrict__ degree) {
  int e = blockIdx.x * blockDim.x + threadIdx.x;
  if (e >= NEDGES) return;
  int s = ei[e];
  int d = ei[NEDGES + e];
  atomicAdd(&degree[d], 1.0f);          // deterministic: integer-valued sums
  adj[s * NPAD + d] = 1.0f;             // set semantics: all writers write 1.0
  adj[d * NPAD + s] = 1.0f;
}

// In-place: adj buffer -> padded normalized Laplacian (zeros outside 500x500)
__global__ void k_build_lap(float* __restrict__ L, const float* __restrict__ degree) {
  int idx = blockIdx.x * blockDim.x + threadIdx.x;
  if (idx >= NPAD * NPAD) return;
  int i = idx >> 9, j = idx & (NPAD - 1);
  float v = 0.0f;
  if (i < NNODES && j < NNODES) {
    float a  = L[idx];
    float di = 1.0f / sqrtf(fmaxf(degree[i], 1.0f));
    float dj = 1.0f / sqrtf(fmaxf(degree[j], 1.0f));
    v = ((i == j) ? 1.0f : 0.0f) - di * a * dj;
  }
  L[idx] = v;
}

// ---------------- expm helpers ----------------

__global__ void k_scale(const float* __restrict__ L, float* __restrict__ B,
                        const float* __restrict__ scales, int s) {
  int idx = blockIdx.x * blockDim.x + threadIdx.x;
  if (idx >= NPAD * NPAD) return;
  float c = -scales[s] * (1.0f / (float)(1 << SQUARINGS));
  B[idx] = c * L[idx];
}

__global__ void k_init_taylor(const float* __restrict__ B, float* __restrict__ P) {
  int idx = blockIdx.x * blockDim.x + threadIdx.x;
  if (idx >= NPAD * NPAD) return;
  int i = idx >> 9, j = idx & (NPAD - 1);
  P[idx] = B[idx] * (1.0f / (float)TAYLOR_ORDER) + ((i == j) ? 1.0f : 0.0f);
}

// C = alpha * (A @ B) + (addI ? I : 0), all 512x512 row-major.
// 256 threads = 8 waves; one 16x16 WMMA tile per wave.
__global__ __launch_bounds__(256) void k_gemm512(const float* __restrict__ A,
                                                 const float* __restrict__ B,
                                                 float* __restrict__ C,
                                                 float alpha, int addI) {
  int wave = threadIdx.x >> 5;
  int lane = threadIdx.x & 31;
  int tile = blockIdx.x * 8 + wave;   // 1024 tiles total
  int tm = tile >> 5;
  int tn = tile & 31;
  int l15   = lane & 15;
  int khalf = (lane >> 4) << 1;       // 0 or 2
  int arow  = tm * 16 + l15;          // A: M = lane%16
  int bcol  = tn * 16 + l15;          // B/C/D: N = lane%16
  v8f acc = {};
  for (int k0 = 0; k0 < NPAD; k0 += 4) {
    v2f a, b;
    a.x = A[arow * NPAD + k0 + khalf];
    a.y = A[arow * NPAD + k0 + khalf + 1];
    b.x = B[(k0 + khalf) * NPAD + bcol];
    b.y = B[(k0 + khalf + 1) * NPAD + bcol];
    acc = wmma_f32(a, b, acc);
  }
  int rbase = tm * 16 + ((lane >> 4) ? 8 : 0);
#pragma unroll
  for (int r = 0; r < 8; ++r) {
    int orow = rbase + r;
    float v = acc[r] * alpha;
    if (addI && (orow == bcol)) v += 1.0f;
    C[orow * NPAD + bcol] = v;
  }
}

// ---------------- big GEMM: Out[s] = E_s @ X  ----------------
// Block: 256 threads (8 waves), each wave a 16-row strip, 32 cols, all 4 scales.
__global__ __launch_bounds__(256) void k_biggemm(const float* __restrict__ E,
                                                 const float* __restrict__ X,
                                                 float* __restrict__ Out) {
  int wave = threadIdx.x >> 5;
  int lane = threadIdx.x & 31;
  int m0 = blockIdx.y * 128 + wave * 16;   // padded rows 0..511
  int n0 = blockIdx.x * 32;                // feature cols
  int l15   = lane & 15;
  int khalf = (lane >> 4) << 1;
  int arow  = m0 + l15;

  v8f acc[NSCALES][2];
#pragma unroll
  for (int s = 0; s < NSCALES; ++s) {
    acc[s][0] = (v8f){};
    acc[s][1] = (v8f){};
  }

  for (int k0 = 0; k0 < NPAD; k0 += 4) {
    int kr0 = k0 + khalf;
    int kr1 = kr0 + 1;
    // X fragments for two 16-wide column tiles, zero-padded past row 499
    v2f b0, b1;
    b0.x = (kr0 < NNODES) ? X[(size_t)kr0 * NFEAT + n0 + l15] : 0.0f;
    b0.y = (kr1 < NNODES) ? X[(size_t)kr1 * NFEAT + n0 + l15] : 0.0f;
    b1.x = (kr0 < NNODES) ? X[(size_t)kr0 * NFEAT + n0 + 16 + l15] : 0.0f;
    b1.y = (kr1 < NNODES) ? X[(size_t)kr1 * NFEAT + n0 + 16 + l15] : 0.0f;
#pragma unroll
    for (int s = 0; s < NSCALES; ++s) {
      const float* Es = E + (size_t)s * NPAD * NPAD;
      v2f a;
      a.x = Es[arow * NPAD + kr0];
      a.y = Es[arow * NPAD + kr1];
      acc[s][0] = wmma_f32(a, b0, acc[s][0]);
      acc[s][1] = wmma_f32(a, b1, acc[s][1]);
    }
  }

  int rbase = m0 + ((lane >> 4) ? 8 : 0);
#pragma unroll
  for (int s = 0; s < NSCALES; ++s) {
#pragma unroll
    for (int t = 0; t < 2; ++t) {
#pragma unroll
      for (int r = 0; r < 8; ++r) {
        int orow = rbase + r;
        if (orow < NNODES) {
          Out[(size_t)s * NNODES * NFEAT + (size_t)orow * NFEAT + n0 + t * 16 + l15]
              = acc[s][t][r];
        }
      }
    }
  }
}

// ---------------- host orchestration ----------------

extern "C" void kernel_launch(void* const* d_in, const int* in_sizes, int n_in,
                              void* d_out, int out_size, void* d_ws, size_t ws_size,
                              hipStream_t stream) {
  (void)in_sizes; (void)n_in; (void)out_size; (void)ws_size;
  const float* x      = (const float*)d_in[0];
  const int*   ei     = (const int*)d_in[1];
  const float* scales = (const float*)d_in[2];
  float* out = (float*)d_out;
  float* ws  = (float*)d_ws;

  const int MSZ = NPAD * NPAD;
  float* degree = ws;                 // 512 floats
  float* L      = ws + NPAD;          // 512x512 (adjacency, then Laplacian)
  float* Bm     = L  + MSZ;           // scaled Laplacian
  float* P0     = Bm + MSZ;           // Taylor / squaring ping
  float* P1     = P0 + MSZ;           // Taylor / squaring pong
  float* Emats  = P1 + MSZ;           // 4 x 512x512 expm results

  // 1) graph -> padded normalized Laplacian
  int nz = NPAD + MSZ;
  k_zero<<<(nz + 255) / 256, 256, 0, stream>>>(ws, nz);
  k_scatter<<<(NEDGES + 255) / 256, 256, 0, stream>>>(ei, L, degree);
  k_build_lap<<<MSZ / 256, 256, 0, stream>>>(L, degree);

  // 2) E_s = expm(-scales[s] * L) by scaling-and-squaring
  for (int s = 0; s < NSCALES; ++s) {
    k_scale<<<MSZ / 256, 256, 0, stream>>>(L, Bm, scales, s);
    k_init_taylor<<<MSZ / 256, 256, 0, stream>>>(Bm, P0);
    float* pin = P0;
    float* pout = P1;
    for (int k = TAYLOR_ORDER - 1; k >= 1; --k) {   // P <- I + (B @ P)/k
      k_gemm512<<<128, 256, 0, stream>>>(Bm, pin, pout, 1.0f / (float)k, 1);
      float* t = pin; pin = pout; pout = t;
    }
    for (int q = 0; q < SQUARINGS; ++q) {           // P <- P @ P
      float* dst = (q == SQUARINGS - 1) ? (Emats + (size_t)s * MSZ) : pout;
      k_gemm512<<<128, 256, 0, stream>>>(pin, pin, dst, 1.0f, 0);
      float* t = pin; pin = pout; pout = t;
    }
  }

  // 3) wavelets[s] = E_s @ x   (all scales in one kernel, x loaded once/tile)
  dim3 grid(NFEAT / 32, NPAD / 128);
  k_biggemm<<<grid, 256, 0, stream>>>(Emats, x, out);
}